// MOERouter_52836687675415
// MI455X (gfx1250) — compile-verified
//
#include <hip/hip_runtime.h>

// MoE router for MI455X (gfx1250, wave32).
// logits = H[T,D] @ W[E,D]^T + b ; softmax ; top-2 ; renorm ; one-hot mask.
// HBM-bound (134 MB of H @ 23.3 TB/s ~ 5.8us); GEMM via V_WMMA_F32_16X16X4_F32.
// W (128 KB) is staged into LDS by the Tensor Data Mover (8 descriptors, one
// per expert row, so the padded LDS stride is preserved for conflict-free
// ds_load_b64 B-fragment reads).

#define TOK 8192
#define DIM 4096
#define NE  8
#define NK  2

#define WROW 4100   // padded LDS row stride (floats): conflict-free ds_load_b64

typedef float v2f __attribute__((ext_vector_type(2)));
typedef float v8f __attribute__((ext_vector_type(8)));
typedef unsigned int u32x4 __attribute__((ext_vector_type(4)));
typedef int i32x4 __attribute__((ext_vector_type(4)));
typedef int i32x8 __attribute__((ext_vector_type(8)));

__global__ __launch_bounds__(256) void moe_router_wmma(
    const float* __restrict__ H,   // [TOK, DIM]
    const float* __restrict__ W,   // [NE, DIM]
    const float* __restrict__ B,   // [NE]
    float* __restrict__ out)       // logits | weights | sel(as float) | mask
{
    extern __shared__ float smem[];
    float* smemW = smem;                 // NE * WROW floats (staged W)
    float* smemL = smem + NE * WROW;     // 8 waves * 16 tokens * NE logits

    const int tid  = threadIdx.x;
    const int wave = tid >> 5;
    const int lane = tid & 31;

    // ---- stage W [8, 4096] into LDS ----
#if __has_builtin(__builtin_amdgcn_tensor_load_to_lds)
    // TDM path: one D# per expert row -> 16 KB DMA into padded LDS row.
    if (wave == 0) {
        #pragma unroll
        for (int e = 0; e < NE; ++e) {
            // LDS byte offset = low 32 bits of generic pointer (LDS aperture).
            unsigned lds_b = (unsigned)(uintptr_t)(smemW + e * WROW);
            unsigned long long ga =
                (unsigned long long)(uintptr_t)(W + (size_t)e * DIM);
            // D# group 0: count=1 | lds_addr | global_addr | type=2
            u32x4 g0;
            g0[0] = 1u;                                   // count=1, user desc
            g0[1] = lds_b;                                // lds_addr (bytes)
            g0[2] = (unsigned)(ga & 0xffffffffu);         // global_addr[31:0]
            g0[3] = (unsigned)((ga >> 32) & 0x01ffffffu)  // global_addr[56:32]
                  | 0x80000000u;                          // type=2 ("image")
            // D# group 1: data_size=4B, dim0=4096, dim1=1, tile0=4096, tile1=1
            i32x8 g1;
            g1[0] = (int)(2u << 16);                      // data_size=2 (4B)
            g1[1] = (int)(((unsigned)DIM & 0xffffu) << 16);   // tensor_dim0 lo16
            g1[2] = (int)((((unsigned)DIM >> 16) & 0xffffu)   // tensor_dim0 hi16
                  | (1u << 16));                          // tensor_dim1 lo16 = 1
            g1[3] = (int)(((unsigned)DIM & 0xffffu) << 16);   // tile_dim0 = 4096
            g1[4] = 1;                                    // tile_dim1=1, tile_dim2=0
            g1[5] = (int)DIM;                             // tensor_dim0_stride[31:0]
            g1[6] = 0;                                    // stride hi / dim1_stride
            g1[7] = 0;
            i32x4 gz; gz[0] = 0; gz[1] = 0; gz[2] = 0; gz[3] = 0;
#if __clang_major__ >= 23
            i32x8 gz8; for (int q = 0; q < 8; ++q) gz8[q] = 0;
            __builtin_amdgcn_tensor_load_to_lds(g0, g1, gz, gz, gz8, 0);
#else
            __builtin_amdgcn_tensor_load_to_lds(g0, g1, gz, gz, 0);
#endif
        }
        __builtin_amdgcn_s_wait_tensorcnt(0);
    }
#else
    // Fallback: cooperative float4 staging.
    #pragma unroll
    for (int i = 0; i < 32; ++i) {
        int g = (i * 256 + tid) * 4;          // flat element idx (mult of 4)
        int e = g >> 12;                      // / 4096
        int c = g & 4095;
        float4 v = *(const float4*)(W + g);
        *(float4*)(smemW + e * WROW + c) = v;
    }
#endif
    __syncthreads();

    // ---- per-wave 16-token tile, D-loop of V_WMMA_F32_16X16X4_F32 ----
    const int tile = blockIdx.x * 128 + wave * 16;   // first token of this wave
    const int arow = lane & 15;                      // token row in tile
    const int kb   = (lane >> 4) << 1;               // K sub-offset: 0 or 2
    const float* aptr = H + (size_t)(tile + arow) * DIM + kb;
    const float* bptr = smemW + (lane & 7) * WROW + kb; // cols 8..15 duplicate

    const float bias = B[lane & 7];                  // folded into C init
    v8f c;
    #pragma unroll
    for (int j = 0; j < 8; ++j) c[j] = bias;

    #pragma unroll 8
    for (int d = 0; d < DIM; d += 4) {
        v2f a = *(const v2f*)(aptr + d);             // global_load_b64
        v2f b = *(const v2f*)(bptr + d);             // ds_load_b64
        c = __builtin_amdgcn_wmma_f32_16x16x4_f32(
                false, a, false, b, (short)0, c, false, false);
    }

    // ---- output regions (flat f32 concat in reference return order) ----
    float* out_logits = out;                         // [TOK, NE]
    float* out_w      = out + TOK * NE;              // [TOK, NK]
    float* out_sel    = out_w + TOK * NK;            // [TOK, NK] (float values)
    float* out_mask   = out_sel + TOK * NK;          // [NE, NK, TOK]

    // D layout: lane L, vgpr j -> token j + 8*(L/16), expert L%16
    const int n   = lane & 15;
    const int mhi = (lane >> 4) << 3;
    float* lrow = smemL + wave * 128;                // 16 tokens * 8 experts
    #pragma unroll
    for (int j = 0; j < 8; ++j) {
        int tl = j + mhi;
        if (n < NE) {
            out_logits[(size_t)(tile + tl) * NE + n] = c[j];
            lrow[tl * NE + n] = c[j];
        }
    }
    __syncthreads();

    // ---- per-token softmax + top-2, lanes 0..15 of each wave ----
    if (lane < 16) {
        const int t = tile + lane;
        float l[NE], p[NE];
        float m = -__builtin_inff();
        #pragma unroll
        for (int e = 0; e < NE; ++e) {
            l[e] = lrow[lane * NE + e];
            m = fmaxf(m, l[e]);
        }
        float z = 0.0f;
        #pragma unroll
        for (int e = 0; e < NE; ++e) { p[e] = __expf(l[e] - m); z += p[e]; }
        float inv = 1.0f / z;
        #pragma unroll
        for (int e = 0; e < NE; ++e) p[e] *= inv;

        int i1 = 0;                                   // strict > keeps lowest
        #pragma unroll
        for (int e = 1; e < NE; ++e) if (p[e] > p[i1]) i1 = e;
        int i2 = (i1 == 0) ? 1 : 0;
        #pragma unroll
        for (int e = 0; e < NE; ++e)
            if (e != i1 && p[e] > p[i2]) i2 = e;

        float s = p[i1] + p[i2];
        out_w[t * NK + 0]   = p[i1] / s;
        out_w[t * NK + 1]   = p[i2] / s;
        out_sel[t * NK + 0] = (float)i1;
        out_sel[t * NK + 1] = (float)i2;
        #pragma unroll
        for (int e = 0; e < NE; ++e) {               // coalesced across lanes
            out_mask[(size_t)(e * NK + 0) * TOK + t] = (e == i1) ? 1.0f : 0.0f;
            out_mask[(size_t)(e * NK + 1) * TOK + t] = (e == i2) ? 1.0f : 0.0f;
        }
    }
}

extern "C" void kernel_launch(void* const* d_in, const int* in_sizes, int n_in,
                              void* d_out, int out_size, void* d_ws, size_t ws_size,
                              hipStream_t stream) {
    (void)in_sizes; (void)n_in; (void)out_size; (void)d_ws; (void)ws_size;
    const float* H = (const float*)d_in[0];
    const float* W = (const float*)d_in[1];
    const float* B = (const float*)d_in[2];
    float* out = (float*)d_out;

    const size_t shmem = (size_t)(NE * WROW + 8 * 128) * sizeof(float); // ~132 KB
    // Raise dynamic-LDS cap (host-side attribute set; not a stream op, so
    // it is graph-capture safe and deterministic).
    (void)hipFuncSetAttribute((const void*)moe_router_wmma,
                              hipFuncAttributeMaxDynamicSharedMemorySize,
                              (int)shmem);

    dim3 grid(TOK / 128);   // 64 workgroups
    dim3 block(256);        // 8 wave32 per workgroup
    moe_router_wmma<<<grid, block, shmem, stream>>>(H, W, B, out);
}